// MessagePassingLayer_16484084482419
// MI455X (gfx1250) — compile-verified
//
#include <hip/hip_runtime.h>

#define SIZE 128
#define AS_STRIDE 132   // padded row stride (floats) for A staging in LDS

typedef __attribute__((ext_vector_type(2))) float v2f;
typedef __attribute__((ext_vector_type(8))) float v8f;

// ---------------------------------------------------------------------------
// CDNA5 async global->LDS copy (bypasses VGPRs, tracked by ASYNCcnt)
// ---------------------------------------------------------------------------
__device__ __forceinline__ void async_copy_b128(unsigned lds_off, const void* gaddr) {
    asm volatile("global_load_async_to_lds_b128 %0, %1, off"
                 :: "v"(lds_off), "v"((unsigned long long)(uintptr_t)gaddr)
                 : "memory");
}
__device__ __forceinline__ void wait_async0() {
    asm volatile("s_wait_asynccnt 0x0" ::: "memory");
}
// Low 32 bits of the flat shared address == LDS byte offset (aperture layout).
__device__ __forceinline__ unsigned lds_offset(const void* p) {
    return (unsigned)(uintptr_t)p;
}

// ---------------------------------------------------------------------------
// Utility kernels
// ---------------------------------------------------------------------------
__global__ void copy4_kernel(const float4* __restrict__ src, float4* __restrict__ dst, size_t n4) {
    size_t i = (size_t)blockIdx.x * blockDim.x + threadIdx.x;
    size_t stride = (size_t)gridDim.x * blockDim.x;
    for (; i < n4; i += stride) dst[i] = src[i];
}

__global__ void zero4_kernel(float4* __restrict__ dst, size_t n4) {
    size_t i = (size_t)blockIdx.x * blockDim.x + threadIdx.x;
    size_t stride = (size_t)gridDim.x * blockDim.x;
    float4 z = {0.f, 0.f, 0.f, 0.f};
    for (; i < n4; i += stride) dst[i] = z;
}

// One wave (32 lanes) per angle row: 128 floats -> 4 hw f32 atomics per lane.
__global__ __launch_bounds__(256) void scatter_angles_kernel(
    const float* __restrict__ ang, const int* __restrict__ idx,
    float* __restrict__ acc, int n) {
    int wave = (int)((blockIdx.x * blockDim.x + threadIdx.x) >> 5);
    int lane = threadIdx.x & 31;
    if (wave >= n) return;
    int b = idx[wave];
    float4 v = ((const float4*)(ang + (size_t)wave * SIZE))[lane];
    float* base = acc + (size_t)b * SIZE + lane * 4;
    unsafeAtomicAdd(base + 0, v.x);
    unsafeAtomicAdd(base + 1, v.y);
    unsafeAtomicAdd(base + 2, v.z);
    unsafeAtomicAdd(base + 3, v.w);
}

__device__ __forceinline__ float wave_sum(float v) {
#pragma unroll
    for (int off = 16; off > 0; off >>= 1) v += __shfl_xor(v, off, 32);
    return v;
}

// One wave32 per row LayerNorm (4 floats per lane).
__global__ __launch_bounds__(256) void ln_kernel(
    const float* __restrict__ x, const float* __restrict__ w,
    const float* __restrict__ b, float* __restrict__ out, int nrows) {
    int row  = (int)((blockIdx.x * blockDim.x + threadIdx.x) >> 5);
    int lane = threadIdx.x & 31;
    if (row >= nrows) return;
    float4 v = ((const float4*)(x + (size_t)row * SIZE))[lane];
    float mean = wave_sum(v.x + v.y + v.z + v.w) * (1.0f / SIZE);
    float dx = v.x - mean, dy = v.y - mean, dz = v.z - mean, dw = v.w - mean;
    float var = wave_sum(dx * dx + dy * dy + dz * dz + dw * dw) * (1.0f / SIZE);
    float rstd = rsqrtf(var + 1e-5f);
    float4 wv = ((const float4*)w)[lane];
    float4 bv = ((const float4*)b)[lane];
    float4 o;
    o.x = dx * rstd * wv.x + bv.x;
    o.y = dy * rstd * wv.y + bv.y;
    o.z = dz * rstd * wv.z + bv.z;
    o.w = dw * rstd * wv.w + bv.w;
    ((float4*)(out + (size_t)row * SIZE))[lane] = o;
}

// ---------------------------------------------------------------------------
// WMMA tile GEMM: each wave computes a 16x128 tile = (gathered 16x128 A) @ W.
// MODE 0: A = h[src] + bondacc[row];  out: relu + atomic scatter to idx[row*2+1]
// MODE 1: A = h2[i] + h2[j] + pair_x; out: relu + atomic scatter to idx[row*2+0]
// MODE 2: A = rows of A0;             out[row] = A1[row] + acc + bias
// ---------------------------------------------------------------------------
template <int MODE>
__global__ __launch_bounds__(128) void gemm_kernel(
    const float* __restrict__ A0, const float* __restrict__ A1,
    const int* __restrict__ idx, const float* __restrict__ W,
    const float* __restrict__ bias, float* __restrict__ out, int nrows) {
    __shared__ float Ws[SIZE * SIZE];          // 64 KB: full weight matrix
    __shared__ float As[4][16 * AS_STRIDE];    // 33 KB: per-wave A tiles (padded)

    const int tid = threadIdx.x;
    // Async direct-to-LDS copy of W (64 KB): 4096 x b128, 32 per thread.
    {
        const unsigned base = lds_offset(Ws);
        const char* g = (const char*)W;
#pragma unroll 4
        for (int i = tid; i < SIZE * SIZE / 4; i += 128)
            async_copy_b128(base + i * 16, g + (size_t)i * 16);
    }

    const int waveid = tid >> 5;
    const int lane   = tid & 31;
    const int tile   = blockIdx.x * 4 + waveid;   // wave-uniform
    const int ntiles = (nrows + 15) >> 4;

    if (tile < ntiles) {
        // Stage A tile: one float4 per lane per row (b128 load + aligned b128 store)
        float* as = As[waveid];
        const int c0 = lane * 4;
#pragma unroll 1
        for (int m = 0; m < 16; ++m) {
            int row = tile * 16 + m;
            float4 val = {0.f, 0.f, 0.f, 0.f};
            if (row < nrows) {
                if (MODE == 0) {
                    int s = idx[row * 2 + 0];
                    float4 a = *(const float4*)(A0 + (size_t)s * SIZE + c0);
                    float4 b = *(const float4*)(A1 + (size_t)row * SIZE + c0);
                    val.x = a.x + b.x; val.y = a.y + b.y;
                    val.z = a.z + b.z; val.w = a.w + b.w;
                } else if (MODE == 1) {
                    int i0 = idx[row * 2 + 0];
                    int j0 = idx[row * 2 + 1];
                    float4 a = *(const float4*)(A0 + (size_t)i0 * SIZE + c0);
                    float4 b = *(const float4*)(A0 + (size_t)j0 * SIZE + c0);
                    float4 c = *(const float4*)(A1 + (size_t)row * SIZE + c0);
                    val.x = a.x + b.x + c.x; val.y = a.y + b.y + c.y;
                    val.z = a.z + b.z + c.z; val.w = a.w + b.w + c.w;
                } else {
                    val = *(const float4*)(A0 + (size_t)row * SIZE + c0);
                }
            }
            *(float4*)(as + m * AS_STRIDE + c0) = val;   // 528*m + 16*lane: 16B aligned
        }
    }
    wait_async0();      // W resident in LDS (this wave's async ops)
    __syncthreads();    // all waves' W slices + A tiles visible
    if (tile >= ntiles) return;

    const float* as = As[waveid];
    const int hi = lane >> 4;   // 0: K/M low half, 1: high half
    const int lo = lane & 15;

    // 8 column tiles of 16; one v8f accumulator live at a time
    for (int ct = 0; ct < 8; ++ct) {
        v8f acc = {0.f, 0.f, 0.f, 0.f, 0.f, 0.f, 0.f, 0.f};
#pragma unroll
        for (int k0 = 0; k0 < SIZE; k0 += 4) {
            const int kk = k0 + 2 * hi;
            v2f a, b;
            a[0] = as[lo * AS_STRIDE + kk];               // A: row=lo, K=kk
            a[1] = as[lo * AS_STRIDE + kk + 1];           //            K=kk+1
            b[0] = Ws[kk * SIZE + ct * 16 + lo];          // B: K=kk,   N=ct*16+lo
            b[1] = Ws[(kk + 1) * SIZE + ct * 16 + lo];    //    K=kk+1
            acc = __builtin_amdgcn_wmma_f32_16x16x4_f32(
                false, a, false, b, (short)0, acc, false, false);
        }
        const int col = ct * 16 + lo;
#pragma unroll
        for (int v = 0; v < 8; ++v) {
            int m = v + 8 * hi;
            int row = tile * 16 + m;
            if (row < nrows) {
                float val = acc[v];
                if (MODE == 2) {
                    out[(size_t)row * SIZE + col] =
                        A1[(size_t)row * SIZE + col] + val + bias[col];
                } else {
                    val = val > 0.0f ? val : 0.0f;
                    int d = (MODE == 0) ? idx[row * 2 + 1] : idx[row * 2 + 0];
                    unsafeAtomicAdd(&out[(size_t)d * SIZE + col], val);
                }
            }
        }
    }
}

// ---------------------------------------------------------------------------
extern "C" void kernel_launch(void* const* d_in, const int* in_sizes, int n_in,
                              void* d_out, int out_size, void* d_ws, size_t ws_size,
                              hipStream_t stream) {
    const float* x       = (const float*)d_in[0];
    const float* bond_x  = (const float*)d_in[1];
    const float* sc_pair = (const float*)d_in[2];
    const float* angles  = (const float*)d_in[3];
    const float* W_bm    = (const float*)d_in[4];
    const float* W_sc    = (const float*)d_in[5];
    const float* W1      = (const float*)d_in[6];
    const float* b1      = (const float*)d_in[7];
    const float* W2      = (const float*)d_in[8];
    const float* b2      = (const float*)d_in[9];
    const float* ln1w    = (const float*)d_in[10];
    const float* ln1b    = (const float*)d_in[11];
    const float* ln2w    = (const float*)d_in[12];
    const float* ln2b    = (const float*)d_in[13];
    // d_in[14] = mask (unused by reference)
    const int* bond_idx  = (const int*)d_in[15];
    const int* sc_idx    = (const int*)d_in[16];
    const int* ang_idx   = (const int*)d_in[17];

    const int na = in_sizes[0] / SIZE;   // atoms
    const int nb = in_sizes[1] / SIZE;   // bonds
    const int np = in_sizes[2] / SIZE;   // pairs
    const int ng = in_sizes[3] / SIZE;   // angles

    float* ws      = (float*)d_ws;
    float* bondacc = ws;                             // nb*128 (dies after bond GEMM)
    float* h       = ws + (size_t)nb * SIZE;         // na*128 (h, then h2)
    float* agg     = h + (size_t)na * SIZE;          // na*128
    float* xmid    = bondacc;                        // reuse bondacc region (na < nb)

    // 1) bondacc = bond_x
    copy4_kernel<<<2048, 256, 0, stream>>>((const float4*)bond_x, (float4*)bondacc,
                                           (size_t)nb * SIZE / 4);
    // 2) bondacc += segment_sum(angles)
    scatter_angles_kernel<<<(ng + 7) / 8, 256, 0, stream>>>(angles, ang_idx, bondacc, ng);
    // 3) h = LN1(x)
    ln_kernel<<<(na + 7) / 8, 256, 0, stream>>>(x, ln1w, ln1b, h, na);
    // 4) agg = 0
    zero4_kernel<<<2048, 256, 0, stream>>>((float4*)agg, (size_t)na * SIZE / 4);
    // 5) msg = relu((h[src] + bondacc) @ W_bm); agg[dst] += msg
    {
        int nt = (nb + 15) / 16;
        gemm_kernel<0><<<(nt + 3) / 4, 128, 0, stream>>>(h, bondacc, bond_idx, W_bm,
                                                         nullptr, agg, nb);
    }
    // 6) xmid = x + agg @ W1 + b1
    {
        int nt = (na + 15) / 16;
        gemm_kernel<2><<<(nt + 3) / 4, 128, 0, stream>>>(agg, x, nullptr, W1, b1, xmid, na);
    }
    // 7) h2 = LN2(xmid)
    ln_kernel<<<(na + 7) / 8, 256, 0, stream>>>(xmid, ln2w, ln2b, h, na);
    // 8) agg = 0
    zero4_kernel<<<2048, 256, 0, stream>>>((float4*)agg, (size_t)na * SIZE / 4);
    // 9) msg2 = relu((h2[i] + h2[j] + sc_pair) @ W_sc); agg[i] += msg2
    {
        int nt = (np + 15) / 16;
        gemm_kernel<1><<<(nt + 3) / 4, 128, 0, stream>>>(h, sc_pair, sc_idx, W_sc,
                                                         nullptr, agg, np);
    }
    // 10) out = xmid + agg @ W2 + b2
    {
        int nt = (na + 15) / 16;
        gemm_kernel<2><<<(nt + 3) / 4, 128, 0, stream>>>(agg, xmid, nullptr, W2, b2,
                                                         (float*)d_out, na);
    }
}